// Trainer_GCN_74242804679333
// MI455X (gfx1250) — compile-verified
//
#include <hip/hip_runtime.h>
#include <hip/hip_bf16.h>

typedef float v2f __attribute__((ext_vector_type(2)));
typedef float v8f __attribute__((ext_vector_type(8)));

#define NN 100000   // nodes
#define NE 1000000  // edges

// ---------------------------------------------------------------------------
// Native f32 global atomic add (global_atomic_add_f32), avoiding any CAS loop.
// ---------------------------------------------------------------------------
__device__ __forceinline__ void atomAddF32(float* p, float v) {
#if defined(__has_builtin) && __has_builtin(__builtin_amdgcn_global_atomic_fadd_f32)
  __builtin_amdgcn_global_atomic_fadd_f32(p, v);
#else
  unsafeAtomicAdd(p, v);  // HIP: lowers to native FP atomic on AMD
#endif
}

// ---------------------------------------------------------------------------
// Degree / norm kernels
// ---------------------------------------------------------------------------
__global__ void deg_init_kernel(float* __restrict__ deg, int n) {
  int i = blockIdx.x * blockDim.x + threadIdx.x;
  if (i < n) deg[i] = 1.0f;  // self-loop
}

__global__ void deg_scatter_kernel(const int* __restrict__ dst,
                                   float* __restrict__ deg, int nE) {
  int e = blockIdx.x * blockDim.x + threadIdx.x;
  if (e < nE) atomAddF32(&deg[dst[e]], 1.0f);
}

__global__ void deg_rsqrt_kernel(float* __restrict__ deg, int n) {
  int i = blockIdx.x * blockDim.x + threadIdx.x;
  if (i < n) deg[i] = rsqrtf(deg[i]);  // deg >= 1 always (self loop)
}

// ---------------------------------------------------------------------------
// GEMM with WMMA f32 16x16x4: hs[m][n] = (X @ W)[m][n] * dinv[m]; acc = hs
// One wave computes one 16x32 output tile (two independent accumulators
// sharing the A fragment -> halves A traffic, pipelines two WMMAs/iter).
// n0 = (blockIdx.y * blockDim.y + threadIdx.y) * 32.
// M % 16 == 0, K % 4 == 0, N % 32 == 0 at all call sites.
// ---------------------------------------------------------------------------
template <int K, int N>
__global__ __launch_bounds__(128) void gemm_dinv_kernel(
    const float* __restrict__ X, const float* __restrict__ W,
    const float* __restrict__ dinv, float* __restrict__ hs,
    float* __restrict__ acc) {
  const int lane = threadIdx.x;  // 0..31
  const int m0 = blockIdx.x * 16;
  const int n0 = (blockIdx.y * blockDim.y + threadIdx.y) * 32;
  const int half = lane >> 4;  // 0 or 1
  const int lm = lane & 15;

  // A fragment source: row m0+lm, cols k+2*half, k+2*half+1 (contiguous pair)
  const float* __restrict__ arow = X + (size_t)(m0 + lm) * K + 2 * half;
  // B fragment source: rows k+2*half, k+2*half+1, col n0+lm (+16 for tile 1)
  const float* __restrict__ bcol = W + (size_t)(2 * half) * N + n0 + lm;

  v8f c0 = {}, c1 = {};
#pragma unroll 4
  for (int k = 0; k < K; k += 4) {
    v2f a = *(const v2f*)(arow + k);
    v2f b0, b1;
    b0.x = bcol[(size_t)k * N];
    b0.y = bcol[(size_t)k * N + N];
    b1.x = bcol[(size_t)k * N + 16];
    b1.y = bcol[(size_t)k * N + N + 16];
    c0 = __builtin_amdgcn_wmma_f32_16x16x4_f32(false, a, false, b0, (short)0,
                                               c0, false, false);
    c1 = __builtin_amdgcn_wmma_f32_16x16x4_f32(false, a, false, b1, (short)0,
                                               c1, false, false);
  }

  const int col = n0 + lm;
#pragma unroll
  for (int r = 0; r < 8; ++r) {
    int row = m0 + r + 8 * half;  // C/D layout: VGPR r -> M=r (lanes<16), M=8+r
    float d = dinv[row];
    float v0 = c0[r] * d;
    float v1 = c1[r] * d;
    size_t base = (size_t)row * N + col;
    hs[base] = v0;        // scaled features for gather
    acc[base] = v0;       // self-loop contribution seeds acc
    hs[base + 16] = v1;
    acc[base + 16] = v1;
  }
}

// ---------------------------------------------------------------------------
// Edge scatter: acc[dst[e]][f] += hs[src[e]][f]   (F threads per edge)
// ---------------------------------------------------------------------------
template <int F>
__global__ __launch_bounds__(256) void edge_scatter_kernel(
    const int* __restrict__ src, const int* __restrict__ dst,
    const float* __restrict__ hs, float* __restrict__ acc, int nE) {
  long t = (long)blockIdx.x * blockDim.x + threadIdx.x;
  int e = (int)(t / F);
  int f = (int)(t % F);
  if (e >= nE) return;
  atomAddF32(acc + (size_t)dst[e] * F + f, hs[(size_t)src[e] * F + f]);
}

// ---------------------------------------------------------------------------
// out[m][f] = relu(acc[m][f] * dinv[m] + bias[f])
// ---------------------------------------------------------------------------
template <int F>
__global__ __launch_bounds__(256) void finalize_relu_kernel(
    const float* __restrict__ acc, const float* __restrict__ dinv,
    const float* __restrict__ bias, float* __restrict__ out, int n) {
  long t = (long)blockIdx.x * blockDim.x + threadIdx.x;
  int m = (int)(t / F);
  int f = (int)(t % F);
  if (m >= n) return;
  float v = acc[t] * dinv[m] + bias[f];
  out[t] = fmaxf(v, 0.0f);
}

// ---------------------------------------------------------------------------
// Heads: wave (32 lanes) per output row; 64-wide dot via float2 + shfl reduce
// ---------------------------------------------------------------------------
__global__ __launch_bounds__(256) void bias_head_kernel(
    const float* __restrict__ h2, const float* __restrict__ WB,
    const float* __restrict__ bB, float* __restrict__ out, int n) {
  int gid = blockIdx.x * blockDim.x + threadIdx.x;
  int node = gid >> 5;
  int lane = threadIdx.x & 31;
  if (node >= n) return;
  float2 v = ((const float2*)(h2 + (size_t)node * 64))[lane];
  float2 w = ((const float2*)WB)[lane];
  float s = v.x * w.x + v.y * w.y;
#pragma unroll
  for (int off = 16; off > 0; off >>= 1) s += __shfl_xor(s, off, 32);
  if (lane == 0) out[node] = s + bB[0];
}

__global__ __launch_bounds__(256) void weight_head_kernel(
    const float* __restrict__ h2, const int* __restrict__ src,
    const int* __restrict__ dst, const float* __restrict__ WW,
    const float* __restrict__ bW, float* __restrict__ out, int nHalf) {
  int gid = blockIdx.x * blockDim.x + threadIdx.x;
  int j = gid >> 5;
  int lane = threadIdx.x & 31;
  if (j >= nHalf) return;
  int s = src[2 * j];
  int d = dst[2 * j];
  float2 a = ((const float2*)(h2 + (size_t)s * 64))[lane];
  float2 b = ((const float2*)(h2 + (size_t)d * 64))[lane];
  float2 w = ((const float2*)WW)[lane];
  float sum = ((a.x + b.x) * 0.5f) * w.x + ((a.y + b.y) * 0.5f) * w.y;
#pragma unroll
  for (int off = 16; off > 0; off >>= 1) sum += __shfl_xor(sum, off, 32);
  if (lane == 0) out[j] = sum + bW[0];
}

// ---------------------------------------------------------------------------
extern "C" void kernel_launch(void* const* d_in, const int* in_sizes, int n_in,
                              void* d_out, int out_size, void* d_ws,
                              size_t ws_size, hipStream_t stream) {
  (void)in_sizes; (void)n_in; (void)out_size; (void)ws_size;

  const float* x  = (const float*)d_in[0];
  const int*   ei = (const int*)d_in[1];
  const float* W1 = (const float*)d_in[2];
  const float* b1 = (const float*)d_in[3];
  const float* W2 = (const float*)d_in[4];
  const float* b2 = (const float*)d_in[5];
  const float* WB = (const float*)d_in[6];
  const float* bB = (const float*)d_in[7];
  const float* WW = (const float*)d_in[8];
  const float* bW = (const float*)d_in[9];
  const int* src = ei;
  const int* dst = ei + NE;
  float* out = (float*)d_out;  // [weights: 500000][biases: 100000]

  float* ws = (float*)d_ws;
  float* dinv = ws;                          // 100000 floats (deg -> dinv in place)
  float* hs1  = ws + 102400;                 // 100000*256
  float* acc1 = hs1 + (size_t)NN * 256;      // 100000*256
  float* hs2  = acc1 + (size_t)NN * 256;     // 100000*64
  float* acc2 = hs2 + (size_t)NN * 64;       // 100000*64

  // --- degree / normalization ---
  deg_init_kernel<<<(NN + 255) / 256, 256, 0, stream>>>(dinv, NN);
  deg_scatter_kernel<<<(NE + 255) / 256, 256, 0, stream>>>(dst, dinv, NE);
  deg_rsqrt_kernel<<<(NN + 255) / 256, 256, 0, stream>>>(dinv, NN);

  // --- layer 1: h1 = relu(gcn(x, W1, b1)) ---
  // 4 waves * 32 cols = 128 cols per block; grid.y = 256/128 = 2
  gemm_dinv_kernel<500, 256>
      <<<dim3(NN / 16, 2), dim3(32, 4), 0, stream>>>(x, W1, dinv, hs1, acc1);
  {
    long tot = (long)NE * 256;
    edge_scatter_kernel<256>
        <<<(unsigned)((tot + 255) / 256), 256, 0, stream>>>(src, dst, hs1,
                                                            acc1, NE);
    long tf = (long)NN * 256;
    finalize_relu_kernel<256>
        <<<(unsigned)((tf + 255) / 256), 256, 0, stream>>>(acc1, dinv, b1, hs1,
                                                           NN);  // h1 -> hs1
  }

  // --- layer 2: h2 = relu(gcn(h1, W2, b2)) ---
  // 2 waves * 32 cols = 64 cols per block; grid.y = 1
  gemm_dinv_kernel<256, 64>
      <<<dim3(NN / 16, 1), dim3(32, 2), 0, stream>>>(hs1, W2, dinv, hs2, acc2);
  {
    long tot = (long)NE * 64;
    edge_scatter_kernel<64>
        <<<(unsigned)((tot + 255) / 256), 256, 0, stream>>>(src, dst, hs2,
                                                            acc2, NE);
    long tf = (long)NN * 64;
    finalize_relu_kernel<64>
        <<<(unsigned)((tf + 255) / 256), 256, 0, stream>>>(acc2, dinv, b2, hs2,
                                                           NN);  // h2 -> hs2
  }

  // --- heads ---
  bias_head_kernel<<<(NN * 32 + 255) / 256, 256, 0, stream>>>(
      hs2, WB, bB, out + NE / 2, NN);
  weight_head_kernel<<<((NE / 2) * 32 + 255) / 256, 256, 0, stream>>>(
      hs2, src, dst, WW, bW, out, NE / 2);
}